// ScorePredictor_50053548868186
// MI455X (gfx1250) — compile-verified
//
#include <hip/hip_runtime.h>
#include <hip/hip_bf16.h>

typedef __attribute__((ext_vector_type(2))) float v2f;
typedef __attribute__((ext_vector_type(4))) float v4f;
typedef __attribute__((ext_vector_type(8))) float v8f;

#define D_FEAT 128

// One wave computes 16 edge scores as diag(A * B^T) via V_WMMA_F32_16X16X4_F32.
// A = x[src rows] (16x128), B^T supplied as the B matrix (4x16 per K-step).
// K-order is permuted so each lane feeds TWO wmma steps from ONE b128 load:
//   iter i, lane group g = lane>>4 loads floats [8i + 4g, 8i + 4g + 4) of its row.
//   step 0 uses elems {0,1}  (K0/K1 from group 0, K2/K3 from group 1)
//   step 1 uses elems {2,3}
// A-side and B-side use identical addressing, so the reduction is exact fp32 FMA.
__global__ __launch_bounds__(256) void sddmm_wmma_kernel(
    const float* __restrict__ x,
    const int* __restrict__ src,
    const int* __restrict__ dst,
    float* __restrict__ out,
    int n_edges)
{
    const int lane = threadIdx.x & 31;
    const int wave = threadIdx.x >> 5;

    const long long group = (long long)blockIdx.x * (blockDim.x >> 5) + wave;
    const long long e0 = group * 16;
    if (e0 >= n_edges) return;              // whole-wave uniform exit: EXEC stays all-ones

    const int r  = lane & 15;               // edge slot within the 16-edge group
    const int g4 = (lane >> 4) << 2;        // float offset of this lane-group's quad

    long long e = e0 + r;
    if (e >= n_edges) e = n_edges - 1;      // clamp tail: duplicate work, stores are guarded

    const float* __restrict__ arow = x + (long long)src[e] * D_FEAT + g4;
    const float* __restrict__ brow = x + (long long)dst[e] * D_FEAT + g4;

    v8f c = {};
#pragma unroll
    for (int i = 0; i < D_FEAT / 8; ++i) {
        v4f av = *(const v4f*)(arow + 8 * i);   // global_load_b128 (16B/lane, rows hot in WGP$/L2)
        v4f bv = *(const v4f*)(brow + 8 * i);

        v2f a0 = { av[0], av[1] };
        v2f b0 = { bv[0], bv[1] };
        v2f a1 = { av[2], av[3] };
        v2f b1 = { bv[2], bv[3] };

        // 8 args: (neg_a, A, neg_b, B, c_mod, C, reuse_a, reuse_b)
        c = __builtin_amdgcn_wmma_f32_16x16x4_f32(false, a0, false, b0, (short)0, c, false, false);
        c = __builtin_amdgcn_wmma_f32_16x16x4_f32(false, a1, false, b1, (short)0, c, false, false);
    }

    // diag(C): m<8  -> lane m,     vgpr m
    //          m>=8 -> lane m+16,  vgpr m-8
    int idx = -1;
    long long eo = -1;
    if (lane < 8)        { idx = lane;      eo = e0 + lane;      }
    else if (lane >= 24) { idx = lane - 24; eo = e0 + lane - 16; }

    float s = c[0];
#pragma unroll
    for (int i = 1; i < 8; ++i)
        if (idx == i) s = c[i];

    if (idx >= 0 && eo < n_edges)
        out[eo] = s;
}

extern "C" void kernel_launch(void* const* d_in, const int* in_sizes, int n_in,
                              void* d_out, int out_size, void* d_ws, size_t ws_size,
                              hipStream_t stream) {
    const float* x   = (const float*)d_in[0];
    const int*   src = (const int*)d_in[1];
    const int*   dst = (const int*)d_in[2];
    float*       out = (float*)d_out;

    const int n_edges = in_sizes[1];

    // 256 threads = 8 waves; 16 edges per wave => 128 edges per block.
    const int edges_per_block = 8 * 16;
    const int grid = (n_edges + edges_per_block - 1) / edges_per_block;

    sddmm_wmma_kernel<<<grid, 256, 0, stream>>>(x, src, dst, out, n_edges);
}